// TimeCellLSTM_37125697306946
// MI455X (gfx1250) — compile-verified
//
#include <hip/hip_runtime.h>

typedef __attribute__((ext_vector_type(2))) float v2f;
typedef __attribute__((ext_vector_type(8))) float v8f;

#define T_LEN 100
#define BATCH 16384
#define ISZ   3
#define HSZ   20
#define G4    80      // 4*H
#define KDIM  24      // H + I + 1 (bias row)
#define KC    6       // K chunks of 4
#define NT    5       // N tiles of 16 (80/16)
#define GST   82      // padded gate-buffer row stride (words)

__device__ __forceinline__ float fast_sigmoid(float x) {
    return __builtin_amdgcn_rcpf(1.0f + __expf(-x));
}
__device__ __forceinline__ float fast_tanh(float x) {
    // tanh(x) = 1 - 2/(exp(2x)+1)
    return 1.0f - 2.0f * __builtin_amdgcn_rcpf(__expf(2.0f * x) + 1.0f);
}

__global__ void __launch_bounds__(32)
lstm_wmma_kernel(const float* __restrict__ x,
                 const float* __restrict__ W_ih,
                 const float* __restrict__ W_hh,
                 const float* __restrict__ b_ih,
                 const float* __restrict__ b_hh,
                 const float* __restrict__ fc_W,
                 const float* __restrict__ fc_b,
                 float* __restrict__ out)
{
    __shared__ float Wc[KDIM][G4];     // combined [W_hh^T ; W_ih^T ; b_ih+b_hh]
    __shared__ float hBuf[16][HSZ];
    __shared__ float cBuf[16][HSZ];
    __shared__ float gBuf[16][GST];    // 16 rows x 80 gates (padded stride)
    __shared__ float fcw[HSZ];

    const int lane = threadIdx.x;             // 0..31
    const int row0 = blockIdx.x * 16;         // batch tile origin

    // ---- one-time setup: build combined B matrix + zero state ----
    for (int idx = lane; idx < KDIM * G4; idx += 32) {
        int k = idx / G4, g = idx % G4;
        float v;
        if (k < HSZ)            v = W_hh[g * HSZ + k];
        else if (k < HSZ + ISZ) v = W_ih[g * ISZ + (k - HSZ)];
        else                    v = b_ih[g] + b_hh[g];
        Wc[k][g] = v;
    }
    if (lane < HSZ) fcw[lane] = fc_W[lane];
    for (int idx = lane; idx < 16 * HSZ; idx += 32) {
        hBuf[idx / HSZ][idx % HSZ] = 0.0f;
        cBuf[idx / HSZ][idx % HSZ] = 0.0f;
    }
    __syncthreads();

    // ---- load WMMA B fragments into registers (constant over time) ----
    // f32 16x16x4 B layout: VGPR0 = K={0|2}, VGPR1 = K={1|3}; N = lane%16
    const int n   = lane & 15;
    const int khi = (lane >> 4) << 1;     // 0 for lanes 0-15, 2 for lanes 16-31
    v2f bfrag[KC][NT];
    #pragma unroll
    for (int kc = 0; kc < KC; ++kc) {
        #pragma unroll
        for (int nt = 0; nt < NT; ++nt) {
            bfrag[kc][nt].x = Wc[4 * kc + khi    ][nt * 16 + n];
            bfrag[kc][nt].y = Wc[4 * kc + khi + 1][nt * 16 + n];
        }
    }

    const float fcb = fc_b[0];
    const int   m   = lane & 15;          // batch row within tile (A layout M)
    const size_t yBase = (size_t)T_LEN * BATCH * HSZ;

    for (int t = 0; t < T_LEN; ++t) {
        // ---- build A fragments: [h(20) | x(3) | 1], K = 24 ----
        v2f afrag[KC];
        #pragma unroll
        for (int kc = 0; kc < 5; ++kc) {
            afrag[kc].x = hBuf[m][4 * kc + khi];
            afrag[kc].y = hBuf[m][4 * kc + khi + 1];
        }
        {
            const float* xp = x + ((size_t)t * BATCH + row0 + m) * ISZ;
            if (lane < 16) { afrag[5].x = xp[0]; afrag[5].y = xp[1]; }
            else           { afrag[5].x = xp[2]; afrag[5].y = 1.0f;  }
        }
        __syncthreads();

        // ---- gates = A x Wc via fp32 WMMA; stash to LDS in D layout ----
        #pragma unroll
        for (int nt = 0; nt < NT; ++nt) {
            v8f acc = {};
            #pragma unroll
            for (int kc = 0; kc < KC; ++kc) {
                acc = __builtin_amdgcn_wmma_f32_16x16x4_f32(
                        false, afrag[kc], false, bfrag[kc][nt],
                        (short)0, acc, false, false);
            }
            #pragma unroll
            for (int r = 0; r < 8; ++r) {
                int M = (lane < 16) ? r : (r + 8);
                gBuf[M][nt * 16 + n] = acc[r];
            }
        }
        __syncthreads();

        // ---- LSTM cell update: 16 rows x 20 hidden = 320 cells, 10/lane ----
        #pragma unroll
        for (int p = 0; p < 10; ++p) {
            int flat = p * 32 + lane;          // 0..319
            int mm = flat / HSZ, j = flat % HSZ;
            float iv = fast_sigmoid(gBuf[mm][j]);
            float fv = fast_sigmoid(gBuf[mm][HSZ + j]);
            float gv = fast_tanh  (gBuf[mm][2 * HSZ + j]);
            float ov = fast_sigmoid(gBuf[mm][3 * HSZ + j]);
            float c  = fv * cBuf[mm][j] + iv * gv;
            cBuf[mm][j] = c;
            float h  = ov * fast_tanh(c);
            hBuf[mm][j] = h;
            // coalesced: consecutive lanes -> consecutive addresses
            out[((size_t)t * BATCH + row0) * HSZ + flat] = h;
        }
        __syncthreads();

        // ---- y[t][b] = h . fc_W + fc_b (lanes 0..15, one row each) ----
        if (lane < 16) {
            float acc = fcb;
            #pragma unroll
            for (int j = 0; j < HSZ; ++j) acc += hBuf[m][j] * fcw[j];
            out[yBase + (size_t)t * BATCH + row0 + m] = acc;
        }
    }
}

extern "C" void kernel_launch(void* const* d_in, const int* in_sizes, int n_in,
                              void* d_out, int out_size, void* d_ws, size_t ws_size,
                              hipStream_t stream) {
    (void)in_sizes; (void)n_in; (void)out_size; (void)d_ws; (void)ws_size;
    const float* x    = (const float*)d_in[0];
    const float* W_ih = (const float*)d_in[1];
    const float* W_hh = (const float*)d_in[2];
    const float* b_ih = (const float*)d_in[3];
    const float* b_hh = (const float*)d_in[4];
    const float* fc_W = (const float*)d_in[5];
    const float* fc_b = (const float*)d_in[6];
    float* out = (float*)d_out;

    dim3 grid(BATCH / 16);   // 1024 tiles of 16 batch rows
    dim3 block(32);          // one wave32 per tile
    lstm_wmma_kernel<<<grid, block, 0, stream>>>(x, W_ih, W_hh, b_ih, b_hh,
                                                 fc_W, fc_b, out);
}